// MyDetector_31009663877365
// MI455X (gfx1250) — compile-verified
//
#include <hip/hip_runtime.h>
#include <hip/hip_bf16.h>
#include <math.h>
#include <stdint.h>

// ---------------- problem constants (from reference) ----------------
#define NCLS     80
#define KDET     300
#define IOU_THR  0.3f
#define CHANNELS 255           // 3*(5+80)

#define HW13 169               // 13*13
#define HW26 676               // 26*26
#define HW52 2704              // 52*52
#define OFF13 0                // box-index base of scale 13 (3*HW cumulative)
#define OFF26 507
#define OFF52 2535
#define TOTAL_BOX 10647        // 3*(169+676+2704)

#define CAND_CAP 1024          // >>100 sigma above expected valid count (~66)
#define NTHR     256           // 8 wave32 waves on one WGP

// ---------------- CDNA5 TDM availability ----------------
#if defined(__gfx1250__) && __has_builtin(__builtin_amdgcn_tensor_load_to_lds)
#define USE_TDM 1
#endif

typedef unsigned int tdm_v4u __attribute__((ext_vector_type(4)));
typedef int          tdm_v4i __attribute__((ext_vector_type(4)));
typedef int          tdm_v8i __attribute__((ext_vector_type(8)));

// Per-slot NMS state overlays the conf staging buffer after the scan phase.
struct NmsData {
    float box[KDET][5];        // ox, oy, w, h, cls
    int   keep[KDET];
};
union StageU {
    float   stage[TOTAL_BOX];  // 42,588 B conf staging (anchor-major per scale)
    NmsData nms;               // 7,200 B  (reused after scan)
};

__device__ __forceinline__ unsigned lds_byte_offset(const void* p) {
    // LDS aperture: flat addr low 32 bits == workgroup-LDS byte offset (ISA §10.2)
    return (unsigned)(unsigned long long)(uintptr_t)p;
}

#ifdef USE_TDM
// One TDM descriptor loads the 3 anchor conf-planes of one scale:
//   2D tile: dim0 = HW contiguous f32, dim1 = 3 rows, row stride = 85*HW elems.
// D# layout per CDNA5 ISA ch.8 (group0: count/lds/global/type, group1: dims).
__device__ __forceinline__ void tdm_load_conf_planes(const float* gsrc,
                                                     unsigned lds_byte,
                                                     unsigned hw) {
    unsigned long long ga = (unsigned long long)(uintptr_t)gsrc;
    tdm_v4u g0;
    g0.x = 1u;                                               // count=1, user mode
    g0.y = lds_byte;                                         // lds_addr
    g0.z = (unsigned)(ga & 0xFFFFFFFFull);                   // global_addr[31:0]
    g0.w = (unsigned)((ga >> 32) & 0x01FFFFFFull) | (2u << 30); // addr[56:32] | type=2
    tdm_v8i g1;
    g1[0] = (int)(2u << 16);                                 // data_size = 4B
    g1[1] = (int)((hw & 0xFFFFu) << 16);                     // tensor_dim0[15:0]
    g1[2] = (int)(((hw >> 16) & 0xFFFFu) | (3u << 16));      // tensor_dim0[31:16] | tensor_dim1=3
    g1[3] = (int)((hw & 0xFFFFu) << 16);                     // tile_dim0 = hw (<=2704, fits 16b)
    g1[4] = 3;                                               // tile_dim1 = 3, tile_dim2 = 0
    g1[5] = (int)(85u * hw);                                 // tensor_dim0_stride (elems)
    g1[6] = 0;
    g1[7] = 0;
    tdm_v4i gz = {0, 0, 0, 0};
#if __has_include(<hip/amd_detail/amd_gfx1250_TDM.h>)
    // amdgpu-toolchain / clang-23: 6-arg form
    tdm_v8i gz8 = {0, 0, 0, 0, 0, 0, 0, 0};
    __builtin_amdgcn_tensor_load_to_lds(g0, g1, gz, gz, gz8, 0);
#else
    // ROCm 7.2 / clang-22: 5-arg form
    __builtin_amdgcn_tensor_load_to_lds(g0, g1, gz, gz, 0);
#endif
}
#endif

__global__ __launch_bounds__(NTHR)
void yolo_post_kernel(const float* __restrict__ out13,
                      const float* __restrict__ out26,
                      const float* __restrict__ out52,
                      const float* __restrict__ a13,
                      const float* __restrict__ a26,
                      const float* __restrict__ a52,
                      const float* __restrict__ thresh_p,
                      float* __restrict__ dout)
{
    __shared__ StageU su;
    __shared__ float  s_conf[CAND_CAP];
    __shared__ int    s_idx[CAND_CAP];
    __shared__ int    s_count;
    __shared__ int    s_sup;

    const int   n   = blockIdx.x;
    const int   tid = threadIdx.x;
    const float thr = thresh_p[0];

    const float* b13 = out13 + (size_t)n * CHANNELS * HW13;
    const float* b26 = out26 + (size_t)n * CHANNELS * HW26;
    const float* b52 = out52 + (size_t)n * CHANNELS * HW52;

    if (tid == 0) s_count = 0;

    // ---- Phase 1: stage the 9 conf planes (channel a*85) into LDS ----
#ifdef USE_TDM
    if (tid == 0) {
        tdm_load_conf_planes(b13, lds_byte_offset(&su.stage[OFF13]), HW13);
        tdm_load_conf_planes(b26, lds_byte_offset(&su.stage[OFF26]), HW26);
        tdm_load_conf_planes(b52, lds_byte_offset(&su.stage[OFF52]), HW52);
    }
    __builtin_amdgcn_s_wait_tensorcnt(0);   // TDM done before barrier release
#else
    for (int t = tid; t < 3 * HW13; t += NTHR)
        su.stage[OFF13 + t] = b13[(size_t)(t / HW13) * 85 * HW13 + (t % HW13)];
    for (int t = tid; t < 3 * HW26; t += NTHR)
        su.stage[OFF26 + t] = b26[(size_t)(t / HW26) * 85 * HW26 + (t % HW26)];
    for (int t = tid; t < 3 * HW52; t += NTHR)
        su.stage[OFF52 + t] = b52[(size_t)(t / HW52) * 85 * HW52 + (t % HW52)];
#endif
    __syncthreads();

    // ---- Phase 2: threshold scan -> candidate list (conf, global box idx) ----
    for (int t = tid; t < TOTAL_BOX; t += NTHR) {
        float c = su.stage[t];
        if (c > thr) {
            int sbase, hw;
            if (t < OFF26)      { sbase = OFF13; hw = HW13; }
            else if (t < OFF52) { sbase = OFF26; hw = HW26; }
            else                { sbase = OFF52; hw = HW52; }
            int local = t - sbase;
            int a     = local / hw;
            int pos   = local - a * hw;
            int idx   = sbase + pos * 3 + a;     // matches reference (h,w,anchor) order
            int slot  = atomicAdd(&s_count, 1);
            if (slot < CAND_CAP) { s_conf[slot] = c; s_idx[slot] = idx; }
        }
    }
    __syncthreads();

    int V = s_count; if (V > CAND_CAP) V = CAND_CAP;
    for (int t = V + tid; t < CAND_CAP; t += NTHR) {
        s_conf[t] = -INFINITY;                   // sorts to the end
        s_idx[t]  = 0x7FFFFFFF;
    }
    __syncthreads();

    // ---- Phase 3: bitonic sort, key = (conf desc, idx asc) == jax top_k order ----
    for (int k = 2; k <= CAND_CAP; k <<= 1) {
        for (int j = k >> 1; j > 0; j >>= 1) {
            for (int i = tid; i < CAND_CAP; i += NTHR) {
                int l = i ^ j;
                if (l > i) {
                    float ci = s_conf[i], cl = s_conf[l];
                    int   ii = s_idx[i],  il = s_idx[l];
                    bool iworse  = (ci < cl) || (ci == cl && ii > il);
                    bool dirDesc = ((i & k) == 0);
                    if (iworse == dirDesc) {
                        s_conf[i] = cl; s_conf[l] = ci;
                        s_idx[i]  = il; s_idx[l]  = ii;
                    }
                }
            }
            __syncthreads();
        }
    }

    int M = V < KDET ? V : KDET;

    // ---- Phase 4: full decode of only the selected boxes (<=300) ----
    for (int i = tid; i < M; i += NTHR) {
        int idx = s_idx[i];
        const float* base; const float* anch;
        float stride; int W_, HW_, rel;
        if (idx < OFF26)      { rel = idx - OFF13; base = b13; anch = a13; stride = 32.f; W_ = 13; HW_ = HW13; }
        else if (idx < OFF52) { rel = idx - OFF26; base = b26; anch = a26; stride = 16.f; W_ = 26; HW_ = HW26; }
        else                  { rel = idx - OFF52; base = b52; anch = a52; stride = 8.f;  W_ = 52; HW_ = HW52; }
        int pos = rel / 3;
        int a   = rel - pos * 3;
        int wi  = pos % W_;
        int hi  = pos / W_;
        const float* ch = base + (size_t)(a * 85) * HW_ + pos;
        float tx = ch[(size_t)1 * HW_];
        float ty = ch[(size_t)2 * HW_];
        float tw = ch[(size_t)3 * HW_];
        float th = ch[(size_t)4 * HW_];
        float ox = ((float)wi + tx) * stride;
        float oy = ((float)hi + ty) * stride;
        float bw = expf(tw) * anch[a * 2 + 0];
        float bh = expf(th) * anch[a * 2 + 1];
        float best = ch[(size_t)5 * HW_];
        int   bk   = 0;
        for (int k = 1; k < NCLS; ++k) {
            float v = ch[(size_t)(5 + k) * HW_];
            if (v > best) { best = v; bk = k; }   // strict > == argmax first-occurrence
        }
        su.nms.box[i][0] = ox;
        su.nms.box[i][1] = oy;
        su.nms.box[i][2] = bw;
        su.nms.box[i][3] = bh;
        su.nms.box[i][4] = (float)bk;
    }
    for (int i = tid; i < KDET; i += NTHR) su.nms.keep[i] = 0;
    __syncthreads();

    // ---- Phase 5: greedy class-wise NMS over sorted candidates ----
    for (int i = 0; i < M; ++i) {
        if (tid == 0) s_sup = 0;
        __syncthreads();
        float oxi = su.nms.box[i][0], oyi = su.nms.box[i][1];
        float wi_ = su.nms.box[i][2], hi_ = su.nms.box[i][3];
        float ci  = su.nms.box[i][4];
        float x1i = oxi - 0.5f * wi_, y1i = oyi - 0.5f * hi_;
        float x2i = oxi + 0.5f * wi_, y2i = oyi + 0.5f * hi_;
        float areai = (x2i - x1i) * (y2i - y1i);
        for (int j = tid; j < i; j += NTHR) {
            if (su.nms.keep[j] && su.nms.box[j][4] == ci) {
                float oxj = su.nms.box[j][0], oyj = su.nms.box[j][1];
                float wj  = su.nms.box[j][2], hj  = su.nms.box[j][3];
                float x1j = oxj - 0.5f * wj, y1j = oyj - 0.5f * hj;
                float x2j = oxj + 0.5f * wj, y2j = oyj + 0.5f * hj;
                float ix1 = fmaxf(x1i, x1j), iy1 = fmaxf(y1i, y1j);
                float ix2 = fminf(x2i, x2j), iy2 = fminf(y2i, y2j);
                float inter  = fmaxf(ix2 - ix1, 0.f) * fmaxf(iy2 - iy1, 0.f);
                float areaj  = (x2j - x1j) * (y2j - y1j);
                float uni    = areai + areaj - inter;
                if (inter / fmaxf(uni, 1e-9f) > IOU_THR) atomicOr(&s_sup, 1);
            }
        }
        __syncthreads();
        if (tid == 0) su.nms.keep[i] = (s_sup == 0) ? 1 : 0;  // all M are valid by construction
        __syncthreads();
    }

    // ---- Phase 6: emit (KDET x 7) rows; non-kept rows are all-zero ----
    for (int i = tid; i < KDET; i += NTHR) {
        float* row = dout + ((size_t)n * KDET + i) * 7;
        if (i < M && su.nms.keep[i]) {
            row[0] = s_conf[i];
            row[1] = su.nms.box[i][0];
            row[2] = su.nms.box[i][1];
            row[3] = su.nms.box[i][2];
            row[4] = su.nms.box[i][3];
            row[5] = su.nms.box[i][4];
            row[6] = (float)n;
        } else {
            row[0] = 0.f; row[1] = 0.f; row[2] = 0.f; row[3] = 0.f;
            row[4] = 0.f; row[5] = 0.f; row[6] = 0.f;
        }
    }
}

extern "C" void kernel_launch(void* const* d_in, const int* in_sizes, int n_in,
                              void* d_out, int out_size, void* d_ws, size_t ws_size,
                              hipStream_t stream) {
    const float* out13 = (const float*)d_in[0];
    const float* out26 = (const float*)d_in[1];
    const float* out52 = (const float*)d_in[2];
    const float* a13   = (const float*)d_in[3];
    const float* a26   = (const float*)d_in[4];
    const float* a52   = (const float*)d_in[5];
    const float* thr   = (const float*)d_in[6];
    float* dout        = (float*)d_out;

    int N = in_sizes[0] / (CHANNELS * HW13);   // 64
    yolo_post_kernel<<<N, NTHR, 0, stream>>>(out13, out26, out52,
                                             a13, a26, a52, thr, dout);
}